// ECOPO_loss_11553462026768
// MI455X (gfx1250) — compile-verified
//
#include <hip/hip_runtime.h>
#include <stdint.h>
#include <math.h>

// Problem constants (from reference: B=4, S=512, V=50257, K=5)
#define VOCAB 50257
#define ROWS  2048
#define TPB   256
#define TILE  4096                      // floats per LDS tile (16 KB)
#define FULL_TILES 12                   // 12*4096 = 49152 aligned elements via async path
#define REM_BASE (FULL_TILES * TILE)    // 49152
#define KTOP  5

// Low 32 bits of a flat shared-aperture address == LDS byte address
// (ISA 10.2: in the LDS aperture the low 32 bits give the LDS address).
__device__ __forceinline__ unsigned lds_addr32(const void* p) {
  return (unsigned)(uintptr_t)p;
}

// Issue one 16 KB tile with 4x global_load_async_to_lds_b128 per wave
// (GVS form: SGPR base + 32-bit VGPR byte offset; 16B per lane per op).
// No offset:imm — the ISA adds INST_OFFSET to BOTH the LDS and memory address.
__device__ __forceinline__ void issue_tile(uint64_t base16 /*16B-aligned row ptr*/,
                                           int t, float* buf, int tid) {
  const unsigned lbase = lds_addr32(buf) + (unsigned)(tid * 16);
  const unsigned voff  = (unsigned)(t * (TILE * 4)) + (unsigned)(tid * 16);
#pragma unroll
  for (int c = 0; c < 4; ++c) {
    asm volatile("global_load_async_to_lds_b128 %0, %1, %2 th:TH_LOAD_NT"
                 :: "v"(lbase + (unsigned)(c * TPB * 16)),
                    "v"(voff  + (unsigned)(c * TPB * 16)),
                    "s"(base16)
                 : "memory");
  }
}

// Register-resident descending top-5 insert (constant indices only -> stays in VGPRs).
__device__ __forceinline__ void top5_insert(float x, int i, float v[KTOP], int id[KTOP]) {
  if (x <= v[4]) return;
  if (x > v[0])      { v[4]=v[3]; id[4]=id[3]; v[3]=v[2]; id[3]=id[2];
                       v[2]=v[1]; id[2]=id[1]; v[1]=v[0]; id[1]=id[0]; v[0]=x; id[0]=i; }
  else if (x > v[1]) { v[4]=v[3]; id[4]=id[3]; v[3]=v[2]; id[3]=id[2];
                       v[2]=v[1]; id[2]=id[1]; v[1]=x; id[1]=i; }
  else if (x > v[2]) { v[4]=v[3]; id[4]=id[3]; v[3]=v[2]; id[3]=id[2]; v[2]=x; id[2]=i; }
  else if (x > v[3]) { v[4]=v[3]; id[4]=id[3]; v[3]=x; id[3]=i; }
  else               { v[4]=x; id[4]=i; }
}

__device__ __forceinline__ void online_update(float x, float& m, float& d) {
  if (x > m) { d = d * expf(m - x) + 1.0f; m = x; }
  else       { d += expf(x - m); }
}

__global__ void __launch_bounds__(TPB)
ecopo_row_kernel(const int* __restrict__ labels,
                 const float* __restrict__ logits,
                 float* __restrict__ ws) {
  __shared__ float bufA[TILE];
  __shared__ float bufB[TILE];
  __shared__ float s_m[TPB], s_d[TPB];
  __shared__ float s_tv[TPB * KTOP];
  __shared__ int   s_ti[TPB * KTOP];
  __shared__ float s_rv[TPB];
  __shared__ int   s_rk[TPB];
  __shared__ float s_topv[KTOP];
  __shared__ int   s_topi[KTOP];

  const int tid = threadIdx.x;
  const int row = blockIdx.x;
  const float* rowp = logits + (size_t)row * VOCAB;

  // Align the async-streamed region to 16 bytes: rows are only 4B-aligned
  // (V*4 mod 16 == 4), so peel 0..3 leading floats and stream the rest as B128.
  const unsigned mis  = (unsigned)(uintptr_t)rowp & 15u;   // 0,4,8,12
  const int      skip = mis ? (int)((16u - mis) >> 2) : 0; // 0..3 floats
  const uint64_t base16 = (uint64_t)(uintptr_t)(rowp + skip);

  float m = -INFINITY, d = 0.0f;
  float tv[KTOP] = {-INFINITY, -INFINITY, -INFINITY, -INFINITY, -INFINITY};
  int   ti[KTOP] = {-1, -1, -1, -1, -1};

  // Peeled head elements (direct loads, <=3 lanes).
  if (tid < skip) {
    float x = rowp[tid];
    online_update(x, m, d);
    top5_insert(x, tid, tv, ti);
  }

  // ---- double-buffered async streaming over 12 full aligned tiles ----
  issue_tile(base16, 0, bufA, tid);

  for (int t = 0; t < FULL_TILES; ++t) {
    float* cur = (t & 1) ? bufB : bufA;
    if (t + 1 < FULL_TILES) {
      float* nxt = ((t + 1) & 1) ? bufB : bufA;
      issue_tile(base16, t + 1, nxt, tid);
      // 4 asyncs/wave/tile; completions are in-order, so <=4 outstanding
      // guarantees the previous tile has fully landed while the next flies.
      asm volatile("s_wait_asynccnt 4" ::: "memory");
    } else {
      asm volatile("s_wait_asynccnt 0" ::: "memory");
    }
    __syncthreads();   // other waves' async chunks now visible too

    const int base = skip + t * TILE;
#pragma unroll 4
    for (int k = 0; k < 16; ++k) {
      const int j = k * TPB + tid;
      float x = cur[j];
      online_update(x, m, d);
      top5_insert(x, base + j, tv, ti);
    }
    __syncthreads();   // tile consumed; its buffer may be overwritten next iter
  }

  // ---- tail: ~1102-1105 elements via direct global loads ----
  for (int gi = skip + REM_BASE + tid; gi < VOCAB; gi += TPB) {
    float x = rowp[gi];
    online_update(x, m, d);
    top5_insert(x, gi, tv, ti);
  }

  // ---- block reduce (m, d) with online-softmax merge ----
  s_m[tid] = m; s_d[tid] = d;
#pragma unroll
  for (int j = 0; j < KTOP; ++j) { s_tv[tid * KTOP + j] = tv[j]; s_ti[tid * KTOP + j] = ti[j]; }
  __syncthreads();
  for (int s = TPB / 2; s > 0; s >>= 1) {
    if (tid < s) {
      float m1 = s_m[tid], d1 = s_d[tid];
      float m2 = s_m[tid + s], d2 = s_d[tid + s];
      float M = fmaxf(m1, m2);
      s_m[tid] = M;
      s_d[tid] = d1 * expf(m1 - M) + d2 * expf(m2 - M);
    }
    __syncthreads();
  }

  // ---- 5-round tournament: global top-5 from per-thread top-5 heads ----
  int p = 0;
  for (int r = 0; r < KTOP; ++r) {
    s_rv[tid] = (p < KTOP) ? s_tv[tid * KTOP + p] : -INFINITY;
    s_rk[tid] = tid;
    __syncthreads();
    for (int s = TPB / 2; s > 0; s >>= 1) {
      if (tid < s && s_rv[tid + s] > s_rv[tid]) {
        s_rv[tid] = s_rv[tid + s];
        s_rk[tid] = s_rk[tid + s];
      }
      __syncthreads();
    }
    if (tid == s_rk[0]) {
      s_topv[r] = s_tv[tid * KTOP + p];
      s_topi[r] = s_ti[tid * KTOP + p];
      ++p;
    }
    __syncthreads();
  }

  // ---- per-row scalar (thread 0) ----
  if (tid == 0) {
    const float M = s_m[0], D = s_d[0];
    const int label = labels[row];
    const float lab = rowp[label];          // line is hot in cache
    const float pos = expf(lab - M) / D;

    float vec[KTOP + 1];
    int   keep[KTOP + 1];
    vec[0] = pos; keep[0] = 1;
    int n_rest = 0;
    for (int k = 0; k < KTOP; ++k) {
      vec[k + 1]  = expf(s_topv[k] - M) / D;
      keep[k + 1] = (s_topi[k] != label) ? 1 : 0;
      n_rest += keep[k + 1];
    }
    float mx = -INFINITY;
    for (int i = 0; i < KTOP + 1; ++i) if (keep[i]) mx = fmaxf(mx, vec[i]);
    float e[KTOP + 1], sum = 0.0f;
    for (int i = 0; i < KTOP + 1; ++i) { e[i] = keep[i] ? expf(vec[i] - mx) : 0.0f; sum += e[i]; }
    const float v0 = e[0] / sum;
    float rest = 0.0f;
    for (int i = 1; i < KTOP + 1; ++i) rest += e[i] / sum;
    const float per_pos = rest / (float)n_rest - v0;

    const int active = (label != 0) && (s_topi[0] != label);
    ws[row]        = active ? per_pos : 0.0f;   // unconditional write => replay-safe
    ws[ROWS + row] = active ? 1.0f    : 0.0f;
  }
}

// Deterministic fixed-order final reduction: 2048 partials -> scalar.
__global__ void __launch_bounds__(TPB)
ecopo_finalize_kernel(const float* __restrict__ ws, float* __restrict__ out) {
  __shared__ float st[TPB], sc[TPB];
  const int tid = threadIdx.x;
  float t = 0.0f, c = 0.0f;
  for (int i = tid; i < ROWS; i += TPB) { t += ws[i]; c += ws[ROWS + i]; }
  st[tid] = t; sc[tid] = c;
  __syncthreads();
  for (int s = TPB / 2; s > 0; s >>= 1) {
    if (tid < s) { st[tid] += st[tid + s]; sc[tid] += sc[tid + s]; }
    __syncthreads();
  }
  if (tid == 0) out[0] = (sc[0] > 0.0f) ? (st[0] / sc[0]) : 0.0f;
}

extern "C" void kernel_launch(void* const* d_in, const int* in_sizes, int n_in,
                              void* d_out, int out_size, void* d_ws, size_t ws_size,
                              hipStream_t stream) {
  const int*   labels = (const int*)d_in[0];    // label_ids: int32 [B,S]
  const float* logits = (const float*)d_in[1];  // logits: float32 [B,S,V]
  float* out = (float*)d_out;                   // scalar float32
  float* ws  = (float*)d_ws;                    // 2*ROWS floats (16 KB)

  ecopo_row_kernel<<<ROWS, TPB, 0, stream>>>(labels, logits, ws);
  ecopo_finalize_kernel<<<1, TPB, 0, stream>>>(ws, out);
}